// Pretrain_15401752724060
// MI455X (gfx1250) — compile-verified
//
#include <hip/hip_runtime.h>
#include <hip/hip_bf16.h>

#define DIM 64
#define PATHS 3
// LDS layout: W stored k-pair interleaved. k2 = k/2 in [0,32), stride padded to
// 160 floats so half-waves (k2 vs k2+1) hit disjoint bank halves (160 % 64 == 32).
#define K2STRIDE 160

typedef __attribute__((ext_vector_type(2))) float v2f;
typedef __attribute__((ext_vector_type(8))) float v8f;

// ---------------------------------------------------------------------------
// XW[p] = table @ W[p]   (fp32, exact, via V_WMMA_F32_16X16X4_F32)
// block = 128 threads = 4 waves; each wave owns a 16(M) x 64(N) output tile.
// W[p] staged in LDS interleaved so every B fragment is one ds_load_b64.
// ---------------------------------------------------------------------------
__global__ __launch_bounds__(128)
void gemm_xw_kernel(const float* __restrict__ table,
                    const float* __restrict__ W,
                    float* __restrict__ XW, int n) {
    __shared__ float sWi[32 * K2STRIDE];   // 20 KB

    const int p = blockIdx.y;
    const float* Wp = W + (size_t)p * DIM * DIM;
    // interleaved stage: element W[k][col] -> sWi[(k/2)*K2STRIDE + col*2 + (k&1)]
    for (int i = threadIdx.x; i < DIM * DIM; i += 128) {
        const int k   = i >> 6;
        const int col = i & 63;
        sWi[(k >> 1) * K2STRIDE + col * 2 + (k & 1)] = Wp[i];
    }
    __syncthreads();

    const int lane = threadIdx.x & 31;   // wave32
    const int wave = threadIdx.x >> 5;
    const int half = lane >> 4;          // 0: lanes 0-15, 1: lanes 16-31
    const int l15  = lane & 15;

    int mbase = (blockIdx.x * 4 + wave) * 16;
    if (mbase + 16 > n) mbase = n - 16;  // n is a multiple of 16: keep EXEC all-1s

    const float* arow = table + (size_t)(mbase + l15) * DIM;

    v8f c0 = {}, c1 = {}, c2 = {}, c3 = {};

#pragma unroll
    for (int k4 = 0; k4 < DIM; k4 += 4) {
        const int k0 = k4 + 2 * half;
        // A 16x4 f32 fragment: lane holds A[mbase+l15][k0], A[..][k0+1]
        v2f a = *(const v2f*)(arow + k0);
        // B 4x16 f32 fragments: one aligned ds_load_b64 each, conflict-free
        const float* bb = sWi + ((k4 >> 1) + half) * K2STRIDE + l15 * 2;
        v2f b0 = *(const v2f*)(bb +  0);
        v2f b1 = *(const v2f*)(bb + 32);
        v2f b2 = *(const v2f*)(bb + 64);
        v2f b3 = *(const v2f*)(bb + 96);
        c0 = __builtin_amdgcn_wmma_f32_16x16x4_f32(false, a, false, b0, (short)0, c0, false, false);
        c1 = __builtin_amdgcn_wmma_f32_16x16x4_f32(false, a, false, b1, (short)0, c1, false, false);
        c2 = __builtin_amdgcn_wmma_f32_16x16x4_f32(false, a, false, b2, (short)0, c2, false, false);
        c3 = __builtin_amdgcn_wmma_f32_16x16x4_f32(false, a, false, b3, (short)0, c3, false, false);
    }

    // C/D layout: VGPR v -> row (v + 8*half), lane l15 -> column within tile
    float* out = XW + ((size_t)p * n + mbase) * DIM;
#pragma unroll
    for (int v = 0; v < 8; ++v) {
        float* o = out + (size_t)(v + 8 * half) * DIM + l15;
        o[0]  = c0[v];
        o[16] = c1[v];
        o[32] = c2[v];
        o[48] = c3[v];
    }
}

// ---------------------------------------------------------------------------
// H[p][r] += val * XW[p][c]  over COO edges.  16 lanes per edge, float4 each.
// Atomic f32 adds resolve in L2 (XW/H are L2-resident).
// ---------------------------------------------------------------------------
__global__ __launch_bounds__(256)
void spmm_kernel(const int* __restrict__ rows, const int* __restrict__ cols,
                 const float* __restrict__ vals, const float* __restrict__ XW,
                 float* __restrict__ H, int nnz, int n) {
    const int p = blockIdx.y;
    const unsigned tid = blockIdx.x * 256u + threadIdx.x;
    const unsigned e   = tid >> 4;
    const unsigned sub = tid & 15u;
    if (e >= (unsigned)nnz) return;

    const size_t pb = (size_t)p * nnz;
    const int   r = rows[pb + e];
    const int   c = cols[pb + e];
    const float v = vals[pb + e];

    const float4 x = *(const float4*)(XW + ((size_t)p * n + c) * DIM + sub * 4);
    float* hp = H + ((size_t)p * n + r) * DIM + sub * 4;
    atomicAdd(hp + 0, v * x.x);
    atomicAdd(hp + 1, v * x.y);
    atomicAdd(hp + 2, v * x.z);
    atomicAdd(hp + 3, v * x.w);
}

// ---------------------------------------------------------------------------
// emb = relu( sum_p wb[p] * relu(H[p]) )
// ---------------------------------------------------------------------------
__global__ __launch_bounds__(256)
void combine_kernel(const float* __restrict__ H, const float* __restrict__ wb,
                    float* __restrict__ emb, int n) {
    const size_t i = (size_t)blockIdx.x * 256 + threadIdx.x;
    const size_t total = (size_t)n * DIM;
    if (i >= total) return;
    const float w0 = wb[0], w1 = wb[1], w2 = wb[2];
    const float h0 = fmaxf(H[i], 0.f);
    const float h1 = fmaxf(H[i + total], 0.f);
    const float h2 = fmaxf(H[i + 2 * total], 0.f);
    emb[i] = fmaxf(w0 * h0 + w1 * h1 + w2 * h2, 0.f);
}

// ---------------------------------------------------------------------------
// out = concat(userEmb[uidx], itemEmb[iidx], itemEmb[nidx])  flat
// ---------------------------------------------------------------------------
__global__ __launch_bounds__(256)
void gather_kernel(const float* __restrict__ uE, const float* __restrict__ iE,
                   const int* __restrict__ uidx, const int* __restrict__ iidx,
                   const int* __restrict__ nidx, float* __restrict__ out, int B) {
    const int t = blockIdx.x * 256 + threadIdx.x;
    const int sect = B * DIM;
    if (t >= 3 * sect) return;
    const int sec = t / sect;
    const int r   = t - sec * sect;
    const int i   = r >> 6;
    const int d   = r & 63;
    int idx;
    const float* src;
    if (sec == 0)      { idx = uidx[i]; src = uE; }
    else if (sec == 1) { idx = iidx[i]; src = iE; }
    else               { idx = nidx[i]; src = iE; }
    out[t] = src[(size_t)idx * DIM + d];
}

extern "C" void kernel_launch(void* const* d_in, const int* in_sizes, int n_in,
                              void* d_out, int out_size, void* d_ws, size_t ws_size,
                              hipStream_t stream) {
    const float* user_table = (const float*)d_in[0];
    const float* item_table = (const float*)d_in[1];
    const float* W_u        = (const float*)d_in[2];
    const float* W_i        = (const float*)d_in[3];
    const float* wb1        = (const float*)d_in[4];
    const float* wb2        = (const float*)d_in[5];
    const float* user_vals  = (const float*)d_in[6];
    const float* item_vals  = (const float*)d_in[7];
    const int*   user_rows  = (const int*)d_in[8];
    const int*   user_cols  = (const int*)d_in[9];
    const int*   item_rows  = (const int*)d_in[10];
    const int*   item_cols  = (const int*)d_in[11];
    const int*   user_idx   = (const int*)d_in[12];
    const int*   item_idx   = (const int*)d_in[13];
    const int*   neg_idx    = (const int*)d_in[14];

    const int nU   = in_sizes[0] / DIM;       // 100000
    const int nI   = in_sizes[1] / DIM;       // 50000
    const int nnzU = in_sizes[8] / PATHS;     // 1.6M
    const int nnzI = in_sizes[10] / PATHS;    // 0.8M
    const int Bsz  = in_sizes[12];            // 8192

    float* ws = (float*)d_ws;
    const size_t uXW = 0;
    const size_t uH  = uXW + (size_t)PATHS * nU * DIM;
    const size_t iXW = uH  + (size_t)PATHS * nU * DIM;
    const size_t iH  = iXW + (size_t)PATHS * nI * DIM;
    const size_t uE  = iH  + (size_t)PATHS * nI * DIM;
    const size_t iE  = uE  + (size_t)nU * DIM;

    // zero scatter targets (graph-capture legal; done every call -> deterministic)
    hipMemsetAsync(ws + uH, 0, (size_t)PATHS * nU * DIM * sizeof(float), stream);
    hipMemsetAsync(ws + iH, 0, (size_t)PATHS * nI * DIM * sizeof(float), stream);

    // per-path GEMMs (WMMA fp32)
    dim3 gU((nU / 16 + 3) / 4, PATHS);
    gemm_xw_kernel<<<gU, 128, 0, stream>>>(user_table, W_u, ws + uXW, nU);
    dim3 gI((nI / 16 + 3) / 4, PATHS);
    gemm_xw_kernel<<<gI, 128, 0, stream>>>(item_table, W_i, ws + iXW, nI);

    // SpMM scatter
    dim3 sU((unsigned)(((long long)nnzU * 16 + 255) / 256), PATHS);
    spmm_kernel<<<sU, 256, 0, stream>>>(user_rows, user_cols, user_vals,
                                        ws + uXW, ws + uH, nnzU, nU);
    dim3 sI((unsigned)(((long long)nnzI * 16 + 255) / 256), PATHS);
    spmm_kernel<<<sI, 256, 0, stream>>>(item_rows, item_cols, item_vals,
                                        ws + iXW, ws + iH, nnzI, nI);

    // relu -> weighted path sum -> relu
    combine_kernel<<<(unsigned)(((size_t)nU * DIM + 255) / 256), 256, 0, stream>>>(
        ws + uH, wb1, ws + uE, nU);
    combine_kernel<<<(unsigned)(((size_t)nI * DIM + 255) / 256), 256, 0, stream>>>(
        ws + iH, wb2, ws + iE, nI);

    // batch gather into concatenated output
    gather_kernel<<<(3 * Bsz * DIM + 255) / 256, 256, 0, stream>>>(
        ws + uE, ws + iE, user_idx, item_idx, neg_idx, (float*)d_out, Bsz);
}